// ExtractionLayer_70970039599504
// MI455X (gfx1250) — compile-verified
//
#include <hip/hip_runtime.h>
#include <hip/hip_bf16.h>

#ifndef __has_builtin
#define __has_builtin(x) 0
#endif

// gfx1250 async global->LDS path (probe-confirmed signature)
#if defined(__HIP_DEVICE_COMPILE__) && defined(__gfx1250__) && \
    __has_builtin(__builtin_amdgcn_global_load_async_to_lds_b128)
#define USE_ASYNC_LDS 1
#else
#define USE_ASYNC_LDS 0
#endif

// gfx1250 f32 WMMA for wave-level sum reduction (guarded, shuffle fallback)
#if defined(__HIP_DEVICE_COMPILE__) && defined(__gfx1250__) && \
    __has_builtin(__builtin_amdgcn_wmma_f32_16x16x4_f32)
#define USE_WMMA_REDUCE 1
#else
#define USE_WMMA_REDUCE 0
#endif

namespace {
constexpr int kB = 64, kV = 64, kF = 32, kL = 16, kP = 64;
// exp(-0.5*(d/0.1)^2) = 2^(-(K*d)^2), K = sqrt(50*log2(e))
constexpr float kScale = 8.4932180459827046f;
constexpr float kLog2e = 1.4426950408889634f;
}  // namespace

typedef float v2f __attribute__((ext_vector_type(2)));
typedef float v8f __attribute__((ext_vector_type(8)));

__device__ __forceinline__ float fast_exp2(float x) {
#if defined(__HIP_DEVICE_COMPILE__) && __has_builtin(__builtin_amdgcn_exp2f)
  return __builtin_amdgcn_exp2f(x);   // v_exp_f32
#else
  return exp2f(x);
#endif
}

// ---------------------------------------------------------------------------
// Kernel 1: amp[f,l,p] = FILTER_LENGTH * softmax(logits[f,l,:], axis=p)
// One wave32 per (f,l) row of 64; block = 128 threads = 4 rows.
// Sum reduction done on the matrix engine: one v_wmma_f32_16x16x4_f32 with
// A = exp partials (64 slots = 2 VGPRs/lane), B = ones -> D rows hold row-sums
// replicated across columns; 7 adds + 1 shuffle finish the 64-value sum.
// ---------------------------------------------------------------------------
__global__ __launch_bounds__(128) void amp_softmax_kernel(
    const float* __restrict__ logits, float* __restrict__ amp) {
  const int lane = threadIdx.x & 31;
  const int wave = threadIdx.x >> 5;
  const int row = blockIdx.x * 4 + wave;  // 0..511 == F*L rows
  const float* src = logits + row * kP;
  float a0 = src[lane];
  float a1 = src[lane + 32];
  float m = fmaxf(a0, a1);
#pragma unroll
  for (int off = 16; off > 0; off >>= 1) m = fmaxf(m, __shfl_xor(m, off, 32));
  float e0 = fast_exp2((a0 - m) * kLog2e);
  float e1 = fast_exp2((a1 - m) * kLog2e);

  float s;
#if USE_WMMA_REDUCE
  {
    v2f a;                    // A: 16x4 f32, 2 VGPRs/lane = all 64 exp values
    a.x = e0;
    a.y = e1;
    v2f ones;                 // B: 4x16 f32 = all ones
    ones.x = 1.0f;
    ones.y = 1.0f;
    v8f c = {0.0f, 0.0f, 0.0f, 0.0f, 0.0f, 0.0f, 0.0f, 0.0f};
    // D[m,n] = sum_k A[m,k]; 8 VGPRs/lane cover rows 0-7 (lanes 0-15) or
    // 8-15 (lanes 16-31); add them, then fold the two halves with one shuffle.
    v8f d = __builtin_amdgcn_wmma_f32_16x16x4_f32(
        /*neg_a=*/false, a, /*neg_b=*/false, ones,
        /*c_mod=*/(short)0, c, /*reuse_a=*/false, /*reuse_b=*/false);
    float h = ((d[0] + d[1]) + (d[2] + d[3])) + ((d[4] + d[5]) + (d[6] + d[7]));
    s = h + __shfl_xor(h, 16, 32);
  }
#else
  s = e0 + e1;
#pragma unroll
  for (int off = 16; off > 0; off >>= 1) s += __shfl_xor(s, off, 32);
#endif

  const float scale = 16.0f / s;  // FILTER_LENGTH / sum
  float* dst = amp + row * kP;
  dst[lane] = e0 * scale;
  dst[lane + 32] = e1 * scale;
}

// ---------------------------------------------------------------------------
// Kernel 2: metric[b,v,f] = sum_{l,p} amp[f,l,p] * 2^(-(K*x[v,f,l]-K*q[b,p])^2)
// Block = 256 threads = 16b x 16v tile for one f. Trans-bound:
// per element 3 VALU (fma, mul, fma-acc) + 1 v_exp_f32.
// ---------------------------------------------------------------------------
__global__ __launch_bounds__(256) void extraction_metric_kernel(
    const float* __restrict__ q2,       // [B,P]
    const float* __restrict__ volumes,  // [V]
    const float* __restrict__ filters,  // [F,L]
    const float* __restrict__ amp,      // [F,L,P]
    float* __restrict__ out) {          // [B,V,F]
  __shared__ float s_q[16 * kP];        // q rows for b-tile      (4 KB)
  __shared__ float s_amp[kL * kP];      // amp slice for this f   (4 KB)
  __shared__ float s_x[16][kL + 1];     // K * vol[v] * filt[f,l] (padded)

  const int t = threadIdx.x;
  const int tx = t & 15;   // v within tile
  const int ty = t >> 4;   // b within tile
  const int v0 = blockIdx.x * 16;
  const int b0 = blockIdx.y * 16;
  const int f = blockIdx.z;

  const float* gq = q2 + b0 * kP;         // 1024 floats = 256 x b128
  const float* ga = amp + f * (kL * kP);  // 1024 floats = 256 x b128

#if USE_ASYNC_LDS
  {
    // Probe-confirmed signature: (global v4i*, shared v4i*, imm, imm)
    typedef int v4i __attribute__((vector_size(16)));
    typedef __attribute__((address_space(1))) v4i* gasv4;
    typedef __attribute__((address_space(3))) v4i* lasv4;
    gasv4 g1 = (gasv4)gq + t;
    gasv4 g2 = (gasv4)ga + t;
    lasv4 l1 = (lasv4)&s_q[0] + t;
    lasv4 l2 = (lasv4)&s_amp[0] + t;
    __builtin_amdgcn_global_load_async_to_lds_b128(g1, l1, 0, 0);
    __builtin_amdgcn_global_load_async_to_lds_b128(g2, l2, 0, 0);
  }
#else
  ((float4*)s_q)[t] = ((const float4*)gq)[t];
  ((float4*)s_amp)[t] = ((const float4*)ga)[t];
#endif

  // Pre-scaled x values: 256 threads cover the 16x16 (v_local, l) grid.
  s_x[ty][tx] = kScale * volumes[v0 + ty] * filters[f * kL + tx];

#if USE_ASYNC_LDS
#if __has_builtin(__builtin_amdgcn_s_wait_asynccnt)
  __builtin_amdgcn_s_wait_asynccnt(0);
#else
  asm volatile("s_wait_asynccnt 0" ::: "memory");
#endif
#endif
  __syncthreads();

  float acc0 = 0.0f, acc1 = 0.0f, acc2 = 0.0f, acc3 = 0.0f;
  const float4* qrow = (const float4*)(s_q + ty * kP);
  for (int l = 0; l < kL; ++l) {
    const float xs = s_x[tx][l];
    const float4* arow = (const float4*)(s_amp + l * kP);
#pragma unroll
    for (int i = 0; i < kP / 4; ++i) {
      float4 q4 = qrow[i];
      float4 a4 = arow[i];
      float d0 = fmaf(-kScale, q4.x, xs);
      float d1 = fmaf(-kScale, q4.y, xs);
      float d2 = fmaf(-kScale, q4.z, xs);
      float d3 = fmaf(-kScale, q4.w, xs);
      float e0 = fast_exp2(-(d0 * d0));
      float e1 = fast_exp2(-(d1 * d1));
      float e2 = fast_exp2(-(d2 * d2));
      float e3 = fast_exp2(-(d3 * d3));
      acc0 = fmaf(a4.x, e0, acc0);
      acc1 = fmaf(a4.y, e1, acc1);
      acc2 = fmaf(a4.z, e2, acc2);
      acc3 = fmaf(a4.w, e3, acc3);
    }
  }
  const float acc = (acc0 + acc1) + (acc2 + acc3);
  out[((size_t)(b0 + ty) * kV + (v0 + tx)) * kF + f] = acc;
}

// ---------------------------------------------------------------------------
extern "C" void kernel_launch(void* const* d_in, const int* in_sizes, int n_in,
                              void* d_out, int out_size, void* d_ws,
                              size_t ws_size, hipStream_t stream) {
  (void)in_sizes; (void)n_in; (void)out_size; (void)ws_size;
  const float* q2      = (const float*)d_in[0];  // [64,64]
  const float* volumes = (const float*)d_in[1];  // [64]
  const float* filters = (const float*)d_in[2];  // [1,32,16]
  const float* logits  = (const float*)d_in[3];  // [1,1,32,16,64]
  float* out = (float*)d_out;                    // [64,64,32] f32
  float* amp = (float*)d_ws;                     // 32768 f32 = 128 KB scratch

  amp_softmax_kernel<<<dim3((kF * kL) / 4), 128, 0, stream>>>(logits, amp);

  dim3 grid(kV / 16, kB / 16, kF);  // (4, 4, 32)
  extraction_metric_kernel<<<grid, 256, 0, stream>>>(q2, volumes, filters, amp,
                                                     out);
}